// ChamferLoss_17841294147741
// MI455X (gfx1250) — compile-verified
//
#include <hip/hip_runtime.h>
#include <math.h>

// Chamfer loss on gfx1250 via V_WMMA_F32_16X16X4_F32.
//
// d2(i,j) = -2 p_i . g_j + |g_j|^2 + |p_i|^2, one WMMA per 16x16 tile:
//   A row i = (px, py, pz, 1)            (16x4, K=4; loop-invariant, 2 VGPRs)
//   B col j = (-2gx, -2gy, -2gz, |g|^2)  (4x16;  one b64 load per lane per tile)
//   C       = |p_i|^2 per row            (loop-invariant, 8 VGPRs)
// Direction-symmetric; launched twice with roles swapped.

typedef __attribute__((ext_vector_type(2))) float v2f;
typedef __attribute__((ext_vector_type(8))) float v8f;

#define CH_EPS   1e-8f
#define CH_N     8192
#define CH_B     4
#define CH_TILE  16

__global__ void chamfer_zero_out(float* out, int n) {
    int i = blockIdx.x * blockDim.x + threadIdx.x;
    if (i < n) out[i] = 0.0f;
}

// src: [B][3][N] coordinate-major. dst: [B*N] float4 = (-2x, -2y, -2z, |v|^2)
__global__ __launch_bounds__(256)
void chamfer_pack(const float* __restrict__ src, float* __restrict__ dst) {
    const int i = blockIdx.x * 256 + threadIdx.x;      // 0 .. B*N-1
    const int b = i / CH_N;
    const int n = i - b * CH_N;
    const float* base = src + (size_t)b * 3 * CH_N;
    const float x = base[n];
    const float y = base[CH_N + n];
    const float z = base[2 * CH_N + n];
    float4 q = make_float4(-2.0f * x, -2.0f * y, -2.0f * z, x * x + y * y + z * z);
    *(float4*)(dst + (size_t)i * 4) = q;
}

// One wave (32 threads) per 16-row strip of 'a'; sweeps all 'b' points.
template <bool PACKED>
__global__ __launch_bounds__(32)
void chamfer_dir_wmma(const float* __restrict__ a,
                      const float* __restrict__ braw,
                      const float* __restrict__ bpk,
                      float* __restrict__ out,
                      float scale) {
    const int strips = CH_N / CH_TILE;          // 512
    const int batch  = blockIdx.x / strips;
    const int ti     = blockIdx.x % strips;
    const int lane   = threadIdx.x;             // 0..31 (wave32)
    const int m      = lane & 15;
    const bool hi    = lane >= 16;

    const float* __restrict__ ax = a + (size_t)batch * 3 * CH_N;
    const float* __restrict__ ay = ax + CH_N;
    const float* __restrict__ az = ay + CH_N;

    // ---- A matrix (16x4 f32): lanes 0-15 hold (K0=px, K1=py),
    //      lanes 16-31 hold (K2=pz, K3=1).
    const int i0 = ti * CH_TILE;
    const float px = ax[i0 + m];
    const float py = ay[i0 + m];
    const float pz = az[i0 + m];
    const float p2 = px * px + py * py + pz * pz;
    v2f A;
    A.x = hi ? pz : px;
    A.y = hi ? 1.0f : py;

    // ---- C matrix: |p|^2 per row, loop-invariant.
    // C/D layout: VGPR v -> row v (lanes 0-15) / row v+8 (lanes 16-31).
    v8f Crow;
#pragma unroll
    for (int v = 0; v < 8; ++v) Crow[v] = __shfl(p2, v + (hi ? 8 : 0), 32);

    // Two running-min accumulators to break the v_min dependency chain.
    v8f acc0, acc1;
#pragma unroll
    for (int v = 0; v < 8; ++v) { acc0[v] = 3.0e38f; acc1[v] = 3.0e38f; }

    // Packed B operand: lane reads exactly its half of the 4x16 B layout
    // ((K0,K1) for lanes 0-15, (K2,K3) for lanes 16-31) -> one b64 per tile.
    const char* bhalf = nullptr;
    const float* __restrict__ bx = nullptr;
    const float* __restrict__ by = nullptr;
    const float* __restrict__ bz = nullptr;
    if (PACKED) {
        bhalf = (const char*)(bpk + (size_t)batch * CH_N * 4) + (hi ? 8 : 0);
    } else {
        bx = braw + (size_t)batch * 3 * CH_N;
        by = bx + CH_N;
        bz = by + CH_N;
    }

    for (int tj = 0; tj < CH_N; tj += 2 * CH_TILE) {
        {
            const int j = tj + m;
            v2f Bm;
            if (PACKED) {
                Bm = *(const v2f*)(bhalf + (size_t)j * 16);
            } else {
                const float gx = bx[j], gy = by[j], gz = bz[j];
                const float g2 = gx * gx + gy * gy + gz * gz;
                Bm.x = hi ? (-2.0f * gz) : (-2.0f * gx);
                Bm.y = hi ? g2           : (-2.0f * gy);
            }
            v8f D = __builtin_amdgcn_wmma_f32_16x16x4_f32(
                false, A, false, Bm, (short)0, Crow, false, false);
#pragma unroll
            for (int v = 0; v < 8; ++v) acc0[v] = fminf(acc0[v], D[v]);
        }
        {
            const int j = tj + CH_TILE + m;
            v2f Bm;
            if (PACKED) {
                Bm = *(const v2f*)(bhalf + (size_t)j * 16);
            } else {
                const float gx = bx[j], gy = by[j], gz = bz[j];
                const float g2 = gx * gx + gy * gy + gz * gz;
                Bm.x = hi ? (-2.0f * gz) : (-2.0f * gx);
                Bm.y = hi ? g2           : (-2.0f * gy);
            }
            v8f D = __builtin_amdgcn_wmma_f32_16x16x4_f32(
                false, A, false, Bm, (short)0, Crow, false, false);
#pragma unroll
            for (int v = 0; v < 8; ++v) acc1[v] = fminf(acc1[v], D[v]);
        }
    }

    v8f acc;
#pragma unroll
    for (int v = 0; v < 8; ++v) acc[v] = fminf(acc0[v], acc1[v]);

    // Min-reduce across the 16 columns within each half-wave.
#pragma unroll
    for (int mask = 1; mask <= 8; mask <<= 1) {
#pragma unroll
        for (int v = 0; v < 8; ++v) {
            float o = __shfl_xor(acc[v], mask, 32);
            acc[v] = fminf(acc[v], o);
        }
    }

    // lanes 0-15 now hold row mins for rows 0..7 (per v), lanes 16-31 rows 8..15.
    float s = 0.0f;
#pragma unroll
    for (int v = 0; v < 8; ++v) s += sqrtf(acc[v] + CH_EPS);
    s += __shfl_xor(s, 16, 32);     // full 16-row strip sum in every lane

    if (lane == 0) atomicAdd(out, s * scale);
}

extern "C" void kernel_launch(void* const* d_in, const int* in_sizes, int n_in,
                              void* d_out, int out_size, void* d_ws, size_t ws_size,
                              hipStream_t stream) {
    const float* pred = (const float*)d_in[0];   // [4,3,8192] f32
    const float* gt   = (const float*)d_in[1];   // [4,3,8192] f32
    float* out = (float*)d_out;                  // scalar f32

    chamfer_zero_out<<<1, 32, 0, stream>>>(out, out_size);

    const int blocks = CH_B * (CH_N / CH_TILE);  // 2048 waves per direction
    const float scale = 1.0f / (float)(CH_B * CH_N);

    const size_t pts      = (size_t)CH_B * CH_N;
    const size_t ws_need  = 2 * pts * 4 * sizeof(float);   // 1 MB of packed float4

    if (ws_size >= ws_need) {
        float* pk_pred = (float*)d_ws;           // [B*N] float4
        float* pk_gt   = pk_pred + pts * 4;      // [B*N] float4
        const int pk_blocks = (int)(pts / 256);
        chamfer_pack<<<pk_blocks, 256, 0, stream>>>(pred, pk_pred);
        chamfer_pack<<<pk_blocks, 256, 0, stream>>>(gt, pk_gt);

        // forward: nearest gt for each predicted point
        chamfer_dir_wmma<true><<<blocks, 32, 0, stream>>>(pred, nullptr, pk_gt, out, scale);
        // backward: nearest predicted point for each gt point
        chamfer_dir_wmma<true><<<blocks, 32, 0, stream>>>(gt, nullptr, pk_pred, out, scale);
    } else {
        chamfer_dir_wmma<false><<<blocks, 32, 0, stream>>>(pred, gt, nullptr, out, scale);
        chamfer_dir_wmma<false><<<blocks, 32, 0, stream>>>(gt, pred, nullptr, out, scale);
    }
}